// RBCombiner_33535104647680
// MI455X (gfx1250) — compile-verified
//
#include <hip/hip_runtime.h>

// RBCombiner backward: fused gather-scale-scatter-add.
//
// Shapes (fixed by the reference):
//   dout_exp : (S1_ROWS + S2_ROWS, 4096) f32   S1_ROWS=16384, S2_ROWS=4096
//   weights  : (S1_ROWS,)            f32
//   indices_s1, indices_s2, s1exp_to_s2 : int32
//   out      : (8192, 4096) f32
//
// Fused math (associativity of +):
//   row i < S1:  out[indices_s1[i] >> 1]              += dout_exp[i]    * weights[indices_s1[i]]
//   row j in S2: out[indices_s1[s1exp_to_s2[j]] >> 1] += dout_exp[S1+j] * weights[indices_s2[j]]
//
// MI455X strategy: out (134 MB) is L2-resident (192 MB L2); atomics resolve in
// L2. Input is streamed once with non-temporal b128 loads so it does not evict
// the output from L2. One 256-thread block (8 wave32 waves) per row.

#define D_MODEL   4096
#define BLOCK     256
#define F4_PER_T  4           // 256 threads * 4 * float4 = 4096 floats = one row

// Native clang vector (not HIP_vector_type) so __builtin_nontemporal_load
// accepts it; 16B aligned -> b128 vmem ops.
typedef float vf4 __attribute__((ext_vector_type(4)));

__global__ __launch_bounds__(BLOCK) void rbc_zero_out(float* __restrict__ out) {
    // One block zeroes one 4096-float output row with b128 stores.
    vf4* p = reinterpret_cast<vf4*>(out + (size_t)blockIdx.x * D_MODEL);
    const vf4 z = {0.f, 0.f, 0.f, 0.f};
#pragma unroll
    for (int v = 0; v < F4_PER_T; ++v)
        p[threadIdx.x + v * BLOCK] = z;
}

__global__ __launch_bounds__(BLOCK) void rbc_scatter_combine(
    const float* __restrict__ dout_exp,     // (S1+S2, 4096)
    const float* __restrict__ weights,      // (S1,)
    const int*   __restrict__ indices_s1,   // (S1,)
    const int*   __restrict__ indices_s2,   // (S2,)
    const int*   __restrict__ s1exp_to_s2,  // (S2,)
    float*       __restrict__ out,          // (n_tokens, 4096)
    int s1_rows)
{
    const int row = blockIdx.x;             // 0 .. S1+S2-1, one row per block

    // Per-row uniform index chain (blockIdx-dependent -> SALU/s_load path).
    int token, widx;
    if (row < s1_rows) {
        const int idx = indices_s1[row];
        widx  = idx;
        token = idx >> 1;                   // TOP_K == 2
    } else {
        const int j = row - s1_rows;
        const int k = s1exp_to_s2[j];
        token = indices_s1[k] >> 1;
        widx  = indices_s2[j];
    }
    const float w = weights[widx];

    const vf4* __restrict__ src =
        reinterpret_cast<const vf4*>(dout_exp + (size_t)row * D_MODEL);
    float* __restrict__ dst = out + (size_t)token * D_MODEL;

#pragma unroll
    for (int v = 0; v < F4_PER_T; ++v) {
        const int e = threadIdx.x + v * BLOCK;      // float4 index in row
        // Streamed-once input: non-temporal (TH=NT) 128-bit load, keeps the
        // L2 reserved for the resident output buffer.
        const vf4 d = __builtin_nontemporal_load(&src[e]);
        const int base = e * 4;
        // Hardware f32 atomic adds; out is L2-resident so these are L2 RMWs.
        unsafeAtomicAdd(&dst[base + 0], d.x * w);
        unsafeAtomicAdd(&dst[base + 1], d.y * w);
        unsafeAtomicAdd(&dst[base + 2], d.z * w);
        unsafeAtomicAdd(&dst[base + 3], d.w * w);
    }
}

extern "C" void kernel_launch(void* const* d_in, const int* in_sizes, int n_in,
                              void* d_out, int out_size, void* d_ws, size_t ws_size,
                              hipStream_t stream) {
    (void)n_in; (void)d_ws; (void)ws_size;

    const float* dout_exp    = (const float*)d_in[0];
    const float* weights     = (const float*)d_in[1];
    const int*   indices_s1  = (const int*)d_in[2];
    const int*   indices_s2  = (const int*)d_in[3];
    const int*   s1exp_to_s2 = (const int*)d_in[4];
    // d_in[5] is n_tokens (device scalar); derive it host-side from out_size.

    float* out = (float*)d_out;

    const int s1_rows  = in_sizes[2];            // 16384
    const int s2_rows  = in_sizes[3];            // 4096
    const int n_tokens = out_size / D_MODEL;     // 8192

    // Pass 1: zero the output (harness poisons d_out). Also warms L2 with the
    // 134 MB output working set, which then stays resident (192 MB L2).
    rbc_zero_out<<<n_tokens, BLOCK, 0, stream>>>(out);

    // Pass 2: fused scatter-combine over all s1 + s2 rows.
    rbc_scatter_combine<<<s1_rows + s2_rows, BLOCK, 0, stream>>>(
        dout_exp, weights, indices_s1, indices_s2, s1exp_to_s2, out, s1_rows);
}